// NaiveCollider_40750649704463
// MI455X (gfx1250) — compile-verified
//
#include <hip/hip_runtime.h>

#define NBODY  4096
#define KBROAD (4 * NBODY)   // 16384
#define KEXACT NBODY         // 4096

typedef float v2f __attribute__((ext_vector_type(2)));
typedef float v8f __attribute__((ext_vector_type(8)));
typedef unsigned int u32x4 __attribute__((ext_vector_type(4)));
typedef int i32x4 __attribute__((ext_vector_type(4)));
typedef int i32x8 __attribute__((ext_vector_type(8)));

// D[M][N] = arow(M) + bcol(N) as a rank-2 product via V_WMMA_F32_16X16X4_F32.
// A[M][:] = [arow, 1, 0, 0]  (lanes 0-15 carry K=0,1; lanes 16-31 carry K=2,3 = 0)
// B[:][N] = [1, bcol, 0, 0]^T (same half-wave K striping)
__device__ __forceinline__ v8f wmma_outer(float arow, float bcol, int half) {
  v2f a, b;
  v8f c = {};
  const bool lo = (half == 0);
  a.x = lo ? arow : 0.0f;  a.y = lo ? 1.0f : 0.0f;
  b.x = lo ? 1.0f : 0.0f;  b.y = lo ? bcol : 0.0f;
  return __builtin_amdgcn_wmma_f32_16x16x4_f32(false, a, false, b,
                                               (short)0, c, false, false);
}

// ---- Tensor Data Mover: 1D contiguous copy of `ndw` dwords global -> LDS ----
// D# built per CDNA5 ISA ch.8: group0 {count=1, lds_addr, global_addr, type=2},
// group1 {data_size=4B, tensor_dim0=W, tensor_dim1=1, tile_dim0=W, tile_dim1=1,
// stride0=W}; groups 2/3 zero (tile_dim2..4 = 0 => unused).
__device__ __forceinline__ void tdm_load_1d(unsigned lds_off, const void* gptr,
                                            unsigned ndw) {
#if __has_builtin(__builtin_amdgcn_tensor_load_to_lds)
  const unsigned long long ga = (unsigned long long)(uintptr_t)gptr;
  u32x4 g0;
  g0[0] = 1u;                                            // count=1, user D#
  g0[1] = lds_off;                                       // LDS byte offset
  g0[2] = (unsigned)(ga & 0xFFFFFFFFu);                  // global addr lo
  g0[3] = (unsigned)((ga >> 32) & 0x1FFFFFFu) | (2u << 30); // addr hi | type=2
  i32x8 g1;
  g1[0] = (int)(2u << 16);                               // data_size = 4 bytes
  g1[1] = (int)((ndw & 0xFFFFu) << 16);                  // tensor_dim0[15:0]
  g1[2] = (int)(((ndw >> 16) & 0xFFFFu) | (1u << 16));   // dim0[31:16]|dim1=1
  g1[3] = (int)((ndw & 0xFFFFu) << 16);                  // dim1 hi=0 | tile_dim0
  g1[4] = 1;                                             // tile_dim1=1, tile_dim2=0
  g1[5] = (int)ndw;                                      // dim0_stride[31:0]
  g1[6] = 0;                                             // stride0 hi | stride1 lo
  g1[7] = 0;
  const i32x4 z4 = {0, 0, 0, 0};
#if defined(__clang_major__) && __clang_major__ >= 23
  const i32x8 z8 = {0, 0, 0, 0, 0, 0, 0, 0};
  __builtin_amdgcn_tensor_load_to_lds(g0, g1, z4, z4, z8, 0);   // clang-23 form
#else
  __builtin_amdgcn_tensor_load_to_lds(g0, g1, z4, z4, 0);       // ROCm 7.2 form
#endif
#else
  (void)lds_off; (void)gptr; (void)ndw;
#endif
}

// Stage all bodies (pos interleaved + radius, 48KB) into LDS via the TDM.
__device__ __forceinline__ void stage_bodies(const float* __restrict__ pos,
                                             const float* __restrict__ rad,
                                             float* spos, float* srad) {
#if __has_builtin(__builtin_amdgcn_tensor_load_to_lds)
  if (threadIdx.x < 32) {            // one wave issues both DMAs
    tdm_load_1d((unsigned)(uintptr_t)spos, pos, 2 * NBODY);
    tdm_load_1d((unsigned)(uintptr_t)srad, rad, NBODY);
    __builtin_amdgcn_s_wait_tensorcnt(0);
  }
  __syncthreads();
#else
  for (int k = threadIdx.x; k < NBODY; k += 256) {
    spos[2 * k] = pos[2 * k]; spos[2 * k + 1] = pos[2 * k + 1]; srad[k] = rad[k];
  }
  __syncthreads();
#endif
}

// ---------------- init workspace (bi/bj/ei/ej = -1, ev = 0, last = -1) -----
__global__ __launch_bounds__(256) void init_ws_kernel(
    int* __restrict__ bi, int* __restrict__ bj,
    int* __restrict__ ei, int* __restrict__ ej, float* __restrict__ ev,
    int* __restrict__ lastI, int* __restrict__ lastJ) {
  const int t = blockIdx.x * 256 + threadIdx.x;
  if (t < KBROAD) { bi[t] = -1; bj[t] = -1; }
  if (t < KEXACT) { ei[t] = -1; ej[t] = -1; ev[2 * t] = 0.0f; ev[2 * t + 1] = 0.0f; }
  if (t < NBODY)  { lastI[t] = -1; lastJ[t] = -1; }
}

// ---------------- pass 1: per-row broad-phase hit counts (WMMA tiles) ------
__global__ __launch_bounds__(256) void broad_count_kernel(
    const float* __restrict__ pos, const float* __restrict__ rad,
    int* __restrict__ rowCount) {
  __shared__ float spos[2 * NBODY];
  __shared__ float srad[NBODY];
  stage_bodies(pos, rad, spos, srad);

  const int lane = threadIdx.x & 31;
  const int half = lane >> 4;
  const int lm   = lane & 15;
  const int band = blockIdx.x * 8 + (threadIdx.x >> 5);  // 256 bands of 16 rows
  const int i0   = band * 16;

  const float xi = spos[2 * (i0 + lm)], yi = spos[2 * (i0 + lm) + 1];
  const float ri = srad[i0 + lm];

  int runLo[8], runHi[8];
#pragma unroll
  for (int v = 0; v < 8; ++v) { runLo[v] = 0; runHi[v] = 0; }

  for (int j0 = 0; j0 < NBODY; j0 += 16) {
    const float xj = spos[2 * (j0 + lm)], yj = spos[2 * (j0 + lm) + 1];
    const float rj = srad[j0 + lm];
    v8f dx = wmma_outer(xi, -xj, half);   // x_i - x_j tile
    v8f dy = wmma_outer(yi, -yj, half);   // y_i - y_j tile
    v8f rs = wmma_outer(ri,  rj, half);   // r_i + r_j tile
#pragma unroll
    for (int v = 0; v < 8; ++v) {
      const int irow = i0 + v + 8 * half;
      const int jcol = j0 + lm;
      const bool hit = (__builtin_fabsf(dx[v]) <= rs[v]) &&
                       (__builtin_fabsf(dy[v]) <= rs[v]) && (irow != jcol);
      const unsigned m = __builtin_amdgcn_ballot_w32(hit);
      runLo[v] += __popc(m & 0xFFFFu);   // row v of tile
      runHi[v] += __popc(m >> 16);       // row v+8 of tile
    }
  }
#pragma unroll
  for (int v = 0; v < 8; ++v) {
    if (lane == v)     rowCount[i0 + v]     = runLo[v];
    if (lane == 8 + v) rowCount[i0 + 8 + v] = runHi[v];
  }
}

// ---------------- pass 2: exclusive scan of 4096 row counts (one block) ----
__global__ __launch_bounds__(256) void scan_rows_kernel(
    const int* __restrict__ rowCount, int* __restrict__ rowOff) {
  __shared__ int s[256];
  const int t = threadIdx.x;
  const int base = t * 16;
  int sum = 0;
  for (int k = 0; k < 16; ++k) sum += rowCount[base + k];
  s[t] = sum;
  __syncthreads();
  for (int off = 1; off < 256; off <<= 1) {
    int v = 0;
    if (t >= off) v = s[t - off];
    __syncthreads();
    s[t] += v;
    __syncthreads();
  }
  int run = (t == 0) ? 0 : s[t - 1];
  for (int k = 0; k < 16; ++k) { rowOff[base + k] = run; run += rowCount[base + k]; }
}

// ---------------- pass 3: re-scan and emit broad pairs at exact ranks ------
__global__ __launch_bounds__(256) void broad_emit_kernel(
    const float* __restrict__ pos, const float* __restrict__ rad,
    const int* __restrict__ rowOff, int* __restrict__ bi, int* __restrict__ bj) {
  __shared__ float spos[2 * NBODY];
  __shared__ float srad[NBODY];
  __builtin_prefetch(rowOff + blockIdx.x * 128, 0, 0);
  stage_bodies(pos, rad, spos, srad);

  const int lane = threadIdx.x & 31;
  const int half = lane >> 4;
  const int lm   = lane & 15;
  const int band = blockIdx.x * 8 + (threadIdx.x >> 5);
  const int i0   = band * 16;

  const float xi = spos[2 * (i0 + lm)], yi = spos[2 * (i0 + lm) + 1];
  const float ri = srad[i0 + lm];
  const int rOffLane = rowOff[i0 + lm];   // lane lm holds rowOff of row i0+lm

  int runLo[8], runHi[8];
#pragma unroll
  for (int v = 0; v < 8; ++v) { runLo[v] = 0; runHi[v] = 0; }

  for (int j0 = 0; j0 < NBODY; j0 += 16) {
    const float xj = spos[2 * (j0 + lm)], yj = spos[2 * (j0 + lm) + 1];
    const float rj = srad[j0 + lm];
    v8f dx = wmma_outer(xi, -xj, half);
    v8f dy = wmma_outer(yi, -yj, half);
    v8f rs = wmma_outer(ri,  rj, half);
#pragma unroll
    for (int v = 0; v < 8; ++v) {
      const int irow = i0 + v + 8 * half;
      const int jcol = j0 + lm;
      const bool hit = (__builtin_fabsf(dx[v]) <= rs[v]) &&
                       (__builtin_fabsf(dy[v]) <= rs[v]) && (irow != jcol);
      const unsigned m = __builtin_amdgcn_ballot_w32(hit);
      const unsigned below = m & ((1u << lane) - 1u);
      const int rank = half ? __popc(below & 0xFFFF0000u) : __popc(below & 0xFFFFu);
      const int roff = __shfl(rOffLane, v + 8 * half, 32);
      const int run  = half ? runHi[v] : runLo[v];
      const int slot = roff + run + rank;   // exact global row-major rank
      if (hit && slot < KBROAD) { bi[slot] = irow; bj[slot] = jcol; }
      runLo[v] += __popc(m & 0xFFFFu);
      runHi[v] += __popc(m >> 16);
    }
  }
}

// ---------------- pass 4: exact phase, ordered compaction of penetrations --
__global__ __launch_bounds__(256) void exact_kernel(
    const float* __restrict__ pos, const float* __restrict__ rad,
    const int* __restrict__ bi, const int* __restrict__ bj,
    int* __restrict__ ei, int* __restrict__ ej, float* __restrict__ ev) {
  __shared__ float spos[2 * NBODY];
  __shared__ float srad[NBODY];
  __shared__ int sc[256];
  stage_bodies(pos, rad, spos, srad);

  const int t = threadIdx.x;
  const int base = t * (KBROAD / 256);   // 64 consecutive slots, in order
  int cnt = 0;
  for (int k = 0; k < KBROAD / 256; ++k) {
    const int i = bi[base + k];
    if (i >= 0) {
      const int j = bj[base + k];
      const float dxv = spos[2 * i] - spos[2 * j];
      const float dyv = spos[2 * i + 1] - spos[2 * j + 1];
      const float dist = __builtin_sqrtf(dxv * dxv + dyv * dyv + 1e-12f);
      if (srad[i] + srad[j] - dist > 0.0f) ++cnt;
    }
  }
  sc[t] = cnt;
  __syncthreads();
  for (int off = 1; off < 256; off <<= 1) {
    int v = 0;
    if (t >= off) v = sc[t - off];
    __syncthreads();
    sc[t] += v;
    __syncthreads();
  }
  int rank = (t == 0) ? 0 : sc[t - 1];
  for (int k = 0; k < KBROAD / 256; ++k) {
    const int i = bi[base + k];
    if (i >= 0) {
      const int j = bj[base + k];
      const float dxv = spos[2 * i] - spos[2 * j];
      const float dyv = spos[2 * i + 1] - spos[2 * j + 1];
      const float dist = __builtin_sqrtf(dxv * dxv + dyv * dyv + 1e-12f);
      const float pen = srad[i] + srad[j] - dist;
      if (pen > 0.0f) {
        if (rank < KEXACT) {
          const float sc2 = pen / dist;
          ei[rank] = i; ej[rank] = j;
          ev[2 * rank] = dxv * sc2; ev[2 * rank + 1] = dyv * sc2;
        }
        ++rank;
      }
    }
  }
}

// ---------------- pass 5a: "last write wins" keys via atomicMax ------------
__global__ __launch_bounds__(256) void scatter_last_kernel(
    const int* __restrict__ ei, const int* __restrict__ ej,
    int* __restrict__ lastI, int* __restrict__ lastJ) {
  const int k = blockIdx.x * 256 + threadIdx.x;
  if (k >= KEXACT) return;
  const int i = ei[k];
  if (i >= 0) atomicMax(&lastI[i], k);
  const int j = ej[k];
  if (j >= 0) atomicMax(&lastJ[j], k);
}

// ---------------- pass 5b: compose output (j-scatter overrides i-scatter) --
__global__ __launch_bounds__(256) void resolve_kernel(
    const float* __restrict__ pos, const int* __restrict__ lastI,
    const int* __restrict__ lastJ, const float* __restrict__ ev,
    float* __restrict__ out) {
  const int b = blockIdx.x * 256 + threadIdx.x;
  if (b >= NBODY) return;
  const float px = pos[2 * b], py = pos[2 * b + 1];
  float ox = px, oy = py;
  const int kI = lastI[b];
  if (kI >= 0) { ox = px + 0.5f * ev[2 * kI]; oy = py + 0.5f * ev[2 * kI + 1]; }
  const int kJ = lastJ[b];
  if (kJ >= 0) { ox = px - 0.5f * ev[2 * kJ]; oy = py - 0.5f * ev[2 * kJ + 1]; }
  out[2 * b] = ox; out[2 * b + 1] = oy;
}

extern "C" void kernel_launch(void* const* d_in, const int* in_sizes, int n_in,
                              void* d_out, int out_size, void* d_ws, size_t ws_size,
                              hipStream_t stream) {
  (void)in_sizes; (void)n_in; (void)out_size; (void)ws_size;
  const float* pos = (const float*)d_in[0];   // [N,2] f32
  const float* rad = (const float*)d_in[1];   // [N]   f32
  float* out = (float*)d_out;                 // [N,2] f32

  int* w = (int*)d_ws;
  int* rowCount = w;                          // 4096
  int* rowOff   = w + NBODY;                  // 4096
  int* bi       = w + 2 * NBODY;              // 16384
  int* bj       = bi + KBROAD;                // 16384
  int* ei       = bj + KBROAD;                // 4096
  int* ej       = ei + KEXACT;                // 4096
  int* lastI    = ej + KEXACT;                // 4096
  int* lastJ    = lastI + NBODY;              // 4096
  float* ev     = (float*)(lastJ + NBODY);    // 8192 floats

  init_ws_kernel<<<KBROAD / 256, 256, 0, stream>>>(bi, bj, ei, ej, ev, lastI, lastJ);
  broad_count_kernel<<<32, 256, 0, stream>>>(pos, rad, rowCount);
  scan_rows_kernel<<<1, 256, 0, stream>>>(rowCount, rowOff);
  broad_emit_kernel<<<32, 256, 0, stream>>>(pos, rad, rowOff, bi, bj);
  exact_kernel<<<1, 256, 0, stream>>>(pos, rad, bi, bj, ei, ej, ev);
  scatter_last_kernel<<<KEXACT / 256, 256, 0, stream>>>(ei, ej, lastI, lastJ);
  resolve_kernel<<<NBODY / 256, 256, 0, stream>>>(pos, lastI, lastJ, ev, out);
}